// AdaptiveTripletLoss_84284438217274
// MI455X (gfx1250) — compile-verified
//
#include <hip/hip_runtime.h>
#include <hip/hip_bf16.h>

typedef __attribute__((ext_vector_type(2))) float v2f;
typedef __attribute__((ext_vector_type(8))) float v8f;

#define B_N   8192
#define D_K   128
#define NCLS  128
#define BIGF  1e9f

// ---------------------------------------------------------------------------
// Kernel 1: per-row L2 norm stats + init of hardest-pos / hardest-neg arrays.
// One wave32 per row (32 lanes x float4 = 128 floats).
// ---------------------------------------------------------------------------
__global__ void __launch_bounds__(256)
atl_norm_kernel(const float* __restrict__ E, float* __restrict__ rnorm,
                float* __restrict__ sq, unsigned* __restrict__ hp,
                unsigned* __restrict__ hn) {
  int row  = (blockIdx.x * blockDim.x + threadIdx.x) >> 5;
  int lane = threadIdx.x & 31;
  if (row >= B_N) return;
  const float4 v = ((const float4*)(E + (size_t)row * D_K))[lane];
  float s = v.x * v.x + v.y * v.y + v.z * v.z + v.w * v.w;
#pragma unroll
  for (int m = 16; m >= 1; m >>= 1) s += __shfl_xor(s, m, 32);
  float nrm = sqrtf(s);
  float r   = 1.0f / fmaxf(nrm, 1e-12f);
  if (lane == 0) {
    rnorm[row] = r;
    sq[row]    = s * r * r;                 // == diag(dot) of normalized rows
    hp[row]    = 0u;                        // bits of +0.0f (distances >= 0)
    hn[row]    = __float_as_uint(BIGF);
  }
}

// ---------------------------------------------------------------------------
// Kernel 2: tiled Gram matrix via V_WMMA_F32_16X16X4_F32, fused with the
// masked hardest-pos (max) / hardest-neg (min) row reductions. The distance
// matrix is never materialized.
//
// K=128 is split across FOUR independent accumulators (dependency distance 4
// between WMMAs writing the same accumulator) so the matrix pipe is not
// serialized on a single C-chain; accumulators are summed in the epilogue.
//
// Fragment layout (ISA 7.12.2, 32-bit A 16x4 / B 4x16): lane l holds, for
// k-step s, elements E[base + (l&15)][4s + 2*(l>>4) + {0,1}] — identical
// addressing for A (row tile) and B (column tile of E^T).
// C/D layout: element (vgpr v, lane l) = D[M][N], M = v + 8*(l>>4), N = l&15.
// ---------------------------------------------------------------------------
__global__ void __launch_bounds__(256)
atl_gram_kernel(const float* __restrict__ E, const int* __restrict__ lab,
                const float* __restrict__ rnorm, const float* __restrict__ sq,
                unsigned* __restrict__ hp, unsigned* __restrict__ hn) {
  const int lane    = threadIdx.x & 31;
  const int wv      = threadIdx.x >> 5;       // wave in workgroup: 0..7
  const int h       = lane >> 4;              // half-wave: 0/1
  const int l16     = lane & 15;
  const int rowbase = blockIdx.y * 128 + wv * 16;
  const int colbase = blockIdx.x * 1024;      // 64 column tiles per block

  // A fragment resident in registers across the whole column sweep.
  const float* rp = E + (size_t)(rowbase + l16) * D_K + 2 * h;
  const float  rr = rnorm[rowbase + l16];
  v2f a[32];
#pragma unroll
  for (int s = 0; s < 32; ++s) {
    float2 t = *(const float2*)(rp + 4 * s);
    a[s].x = t.x * rr;
    a[s].y = t.y * rr;
  }

  int   rl[8];
  float sqr[8];
#pragma unroll
  for (int v = 0; v < 8; ++v) {
    int r  = rowbase + v + 8 * h;             // output-row owned by (v, half)
    rl[v]  = lab[r];
    sqr[v] = sq[r];
  }

  float hpmax[8], hnmin[8];
#pragma unroll
  for (int v = 0; v < 8; ++v) { hpmax[v] = 0.0f; hnmin[v] = BIGF; }

  for (int ct = 0; ct < 64; ++ct) {
    const int    col = colbase + ct * 16 + l16;
    const float* cp  = E + (size_t)col * D_K + 2 * h;
    const float  rc  = rnorm[col];
    const int    cl  = lab[col];
    const float  sqc = sq[col];

    v8f acc[4];
#pragma unroll
    for (int q = 0; q < 4; ++q) acc[q] = (v8f){0.f,0.f,0.f,0.f,0.f,0.f,0.f,0.f};

#pragma unroll
    for (int s = 0; s < 32; ++s) {
      float2 t = *(const float2*)(cp + 4 * s);
      v2f b;
      b.x = t.x * rc;
      b.y = t.y * rc;
      // D = A(16x4,f32) * B(4x16,f32) + C(16x16,f32); round-robin over 4
      // accumulators -> WMMAs on the same accumulator are 4 apart.
      acc[s & 3] = __builtin_amdgcn_wmma_f32_16x16x4_f32(
          /*neg_a=*/false, a[s], /*neg_b=*/false, b,
          /*c_mod=*/(short)0, acc[s & 3], /*reuse_a=*/false, /*reuse_b=*/false);
    }

#pragma unroll
    for (int v = 0; v < 8; ++v) {
      float dot = (acc[0][v] + acc[1][v]) + (acc[2][v] + acc[3][v]);
      float d2  = sqc - 2.0f * dot + sqr[v];
      d2        = fmaxf(d2, 0.0f);
      float d   = (d2 > 0.0f) ? sqrtf(d2) : 0.0f;  // zmask semantics
      if (rl[v] == cl) hpmax[v] = fmaxf(hpmax[v], d);  // diag harmless: d=0
      else             hnmin[v] = fminf(hnmin[v], d);
    }
  }

  // Reduce over the 16 column lanes of each half-wave.
#pragma unroll
  for (int v = 0; v < 8; ++v) {
#pragma unroll
    for (int m = 8; m >= 1; m >>= 1) {
      hpmax[v] = fmaxf(hpmax[v], __shfl_xor(hpmax[v], m, 16));
      hnmin[v] = fminf(hnmin[v], __shfl_xor(hnmin[v], m, 16));
    }
  }
  if (l16 == 0) {
#pragma unroll
    for (int v = 0; v < 8; ++v) {
      int r = rowbase + v + 8 * h;
      // Non-negative floats: uint bit-pattern order == float order.
      atomicMax(&hp[r], __float_as_uint(hpmax[v]));
      atomicMin(&hn[r], __float_as_uint(hnmin[v]));
    }
  }
}

// ---------------------------------------------------------------------------
// Kernel 3: validity from exact class counts, per-row adaptive triplet term,
// deterministic tree reduction, scalar output.
// ---------------------------------------------------------------------------
__global__ void __launch_bounds__(1024)
atl_finalize_kernel(const int* __restrict__ lab, const unsigned* __restrict__ hp,
                    const unsigned* __restrict__ hn, float* __restrict__ out) {
  __shared__ int   cnt[NCLS];
  __shared__ float ssum[1024];
  __shared__ float scnt[1024];
  const int t = threadIdx.x;
  if (t < NCLS) cnt[t] = 0;
  __syncthreads();
  for (int i = t; i < B_N; i += 1024) atomicAdd(&cnt[lab[i]], 1);
  __syncthreads();

  float ls = 0.0f, lc = 0.0f;
  for (int i = t; i < B_N; i += 1024) {
    int  L      = lab[i];
    bool valid  = (cnt[L] >= 2) && (cnt[L] < B_N);
    float hpv   = __uint_as_float(hp[i]);
    float hnv   = __uint_as_float(hn[i]);
    float marg  = 0.5f * (1.0f + hpv);            // MARGIN * (1 + hardest_pos)
    float pr    = fmaxf(hpv - hnv + marg, 0.0f);
    if (valid) { ls += pr; lc += 1.0f; }
  }
  ssum[t] = ls;
  scnt[t] = lc;
  __syncthreads();
#pragma unroll
  for (int off = 512; off >= 1; off >>= 1) {
    if (t < off) { ssum[t] += ssum[t + off]; scnt[t] += scnt[t + off]; }
    __syncthreads();
  }
  if (t == 0) out[0] = ssum[0] / fmaxf(scnt[0], 1.0f);
}

// ---------------------------------------------------------------------------
extern "C" void kernel_launch(void* const* d_in, const int* in_sizes, int n_in,
                              void* d_out, int out_size, void* d_ws, size_t ws_size,
                              hipStream_t stream) {
  const float* E   = (const float*)d_in[0];   // [8192,128] fp32
  const int*   lab = (const int*)d_in[1];     // [8192] int32

  float*    ws    = (float*)d_ws;             // needs 128 KB
  float*    rnorm = ws;
  float*    sq    = ws + B_N;
  unsigned* hp    = (unsigned*)(ws + 2 * B_N);
  unsigned* hn    = (unsigned*)(ws + 3 * B_N);

  // 8192 waves (one per row): 1024 blocks x 256 threads
  atl_norm_kernel<<<dim3(1024), dim3(256), 0, stream>>>(E, rnorm, sq, hp, hn);

  // 8 column chunks x 64 row blocks; 8 waves per block (16 rows each)
  atl_gram_kernel<<<dim3(8, 64), dim3(256), 0, stream>>>(E, lab, rnorm, sq, hp, hn);

  atl_finalize_kernel<<<dim3(1), dim3(1024), 0, stream>>>(lab, hp, hn, (float*)d_out);
}